// Dir_RHS_Rec_17841294147720
// MI455X (gfx1250) — compile-verified
//
#include <hip/hip_runtime.h>

// CDNA5 / gfx1250 (MI455X) Jacobi sweep.
//  - Memory-bound: ~201 MB traffic -> ~8.6 us at 23.3 TB/s. Optimize data path.
//  - Haloed v-tiles staged to LDS via global_load_async_to_lds_b32 (ASYNCcnt).
//  - 5-point stencil evaluated as D = Sv*V + V*Sh + cof*F with
//    V_WMMA_F32_16X16X4_F32 (constant banded shift matrices, K=18 pad to 20).

typedef __attribute__((ext_vector_type(2))) float v2f;
typedef __attribute__((ext_vector_type(8))) float v8f;

#define IMG_H 1024
#define IMG_W 1024
#define TILE_X 64             // interior columns per workgroup
#define TILE_Y 32             // interior rows per workgroup
#define HALO_W (TILE_X + 2)   // 66
#define HALO_H (TILE_Y + 2)   // 34
#define LDS_STRIDE 68         // padded row stride (floats); 68 mod 64 = 4 -> conflict-light

__global__ __launch_bounds__(256) void jacobi_wmma_kernel(
    const float* __restrict__ u, const float* __restrict__ f,
    float* __restrict__ out) {
  __shared__ float tile[HALO_H * LDS_STRIDE];

  const int img = blockIdx.z;
  const int rowBase = 1 + TILE_Y * blockIdx.y;  // first interior row of this WG tile
  const int colBase = 1 + TILE_X * blockIdx.x;  // first interior col
  const int tid = threadIdx.x;
  const size_t imgOff = (size_t)img * (IMG_H * IMG_W);

  // ---------------------------------------------------------------
  // Stage haloed v-tile (u with Dirichlet-zero ring) into LDS.
  // In-range cells: async global->LDS DMA (one DWORD per lane).
  // Boundary / out-of-range cells: plain DS store of 0.
  // ---------------------------------------------------------------
  for (int e = tid; e < HALO_H * HALO_W; e += 256) {
    const int r = e / HALO_W;
    const int c = e - r * HALO_W;
    const int gr = rowBase - 1 + r;
    const int gc = colBase - 1 + c;
    const unsigned ldsOff = (unsigned)((r * LDS_STRIDE + c) * 4);
    const bool inside =
        (gr >= 1) && (gr <= IMG_H - 2) && (gc >= 1) && (gc <= IMG_W - 2);
    if (inside) {
      // GVS mode: mem_addr = SADDR(u) + VADDR(32-bit byte offset)
      const unsigned goff =
          (unsigned)((imgOff + (size_t)gr * IMG_W + (size_t)gc) * 4u);
      asm volatile("global_load_async_to_lds_b32 %0, %1, %2"
                   :
                   : "v"(ldsOff), "v"(goff), "s"(u)
                   : "memory");
    } else {
      tile[r * LDS_STRIDE + c] = 0.0f;  // Dirichlet value
    }
  }
  asm volatile("s_wait_asynccnt 0x0" ::: "memory");  // drain this wave's DMAs
  __syncthreads();                                   // + DScnt + all waves arrived

  // ---------------------------------------------------------------
  // Per-wave 16x16 WMMA tile.
  // ---------------------------------------------------------------
  const int wave = tid >> 5;           // 8 waves: 4 x-subtiles * 2 y-subtiles
  const int lane = tid & 31;
  const int subX = (wave & 3) * 16;    // halo-coords origin of subtile
  const int subY = (wave >> 2) * 16;
  const int mn = lane & 15;            // A: row M ; B/C/D: col N
  const int halfK = (lane < 16) ? 0 : 2;  // K offset carried by this half-wave

  constexpr float COF = -1.0f / (4.0f * 1023.0f * 1023.0f);
  constexpr float WGT = 0.25f;

  // Accumulator = cof * f, laid out per the 16x16 f32 C/D VGPR mapping:
  // vgpr v, lane L -> element (M = v + (L<16?0:8), N = L&15).
  v8f acc;
#pragma unroll
  for (int v = 0; v < 8; ++v) {
    const int m = v + ((lane < 16) ? 0 : 8);
    const int gr = min(rowBase + subY + m, IMG_H - 1);  // clamp (overhang rows)
    const int gc = min(colBase + subX + mn, IMG_W - 1);
    acc[v] = COF * f[imgOff + (size_t)gr * IMG_W + (size_t)gc];
  }

  // K = 18 (halo span), padded to 20 -> 5 slices of K=4.
  // A 16x4 layout: lane L holds (M = L&15, K = k0, k0+1) with k0 = 4*kk + (L<16?0:2).
  // B 4x16 layout: lane L holds (N = L&15, K = k0, k0+1), same k0.
#pragma unroll
  for (int kk = 0; kk < 5; ++kk) {
    const int k0 = kk * 4 + halfK;

    // ---- vertical term: D += Sv(16x20) * Vv(20x16) ----
    {
      v2f a, b;
      a.x = (k0 == mn || k0 == mn + 2) ? WGT : 0.0f;        // Sv[m,k]
      a.y = (k0 + 1 == mn || k0 + 1 == mn + 2) ? WGT : 0.0f;
      const int rb0 = subY + ((k0 < 18) ? k0 : 17);          // clamp padded K:
      const int rb1 = subY + ((k0 + 1 < 18) ? k0 + 1 : 17);  // A=0 there, keep B finite
      b.x = tile[rb0 * LDS_STRIDE + subX + 1 + mn];
      b.y = tile[rb1 * LDS_STRIDE + subX + 1 + mn];
      acc = __builtin_amdgcn_wmma_f32_16x16x4_f32(
          false, a, false, b, (short)0, acc, false, false);
    }

    // ---- horizontal term: D += Vh(16x20) * Sh(20x16) ----
    {
      v2f a, b;
      const int row = subY + 1 + mn;                         // halo row of M=mn
      const int cb0 = subX + ((k0 < 18) ? k0 : 17);
      const int cb1 = subX + ((k0 + 1 < 18) ? k0 + 1 : 17);
      a.x = tile[row * LDS_STRIDE + cb0];
      a.y = tile[row * LDS_STRIDE + cb1];
      b.x = (k0 == mn || k0 == mn + 2) ? WGT : 0.0f;         // Sh[k,n]
      b.y = (k0 + 1 == mn || k0 + 1 == mn + 2) ? WGT : 0.0f;
      acc = __builtin_amdgcn_wmma_f32_16x16x4_f32(
          false, a, false, b, (short)0, acc, false, false);
    }
  }

  // Store interior results (skip overhang past row/col 1022).
#pragma unroll
  for (int v = 0; v < 8; ++v) {
    const int m = v + ((lane < 16) ? 0 : 8);
    const int gr = rowBase + subY + m;
    const int gc = colBase + subX + mn;
    if (gr <= IMG_H - 2 && gc <= IMG_W - 2) {
      out[imgOff + (size_t)gr * IMG_W + (size_t)gc] = acc[v];
    }
  }
}

// Writes the Dirichlet value (0) on the perimeter of every image.
__global__ void dirichlet_boundary_kernel(float* __restrict__ out, int nimg) {
  const int per = 4 * IMG_W;
  const int idx = blockIdx.x * blockDim.x + threadIdx.x;
  if (idx >= nimg * per) return;
  const int img = idx / per;
  const int rem = idx - img * per;
  const int side = rem / IMG_W;
  const int p = rem - side * IMG_W;
  const size_t base = (size_t)img * (IMG_H * IMG_W);
  size_t off;
  if (side == 0)      off = (size_t)p;                                   // row 0
  else if (side == 1) off = (size_t)(IMG_H - 1) * IMG_W + p;             // row 1023
  else if (side == 2) off = (size_t)p * IMG_W;                           // col 0
  else                off = (size_t)p * IMG_W + (IMG_W - 1);             // col 1023
  out[base + off] = 0.0f;
}

extern "C" void kernel_launch(void* const* d_in, const int* in_sizes, int n_in,
                              void* d_out, int out_size, void* d_ws,
                              size_t ws_size, hipStream_t stream) {
  (void)n_in; (void)d_ws; (void)ws_size; (void)out_size;
  const float* u = (const float*)d_in[0];
  const float* f = (const float*)d_in[1];
  // d_in[2] is the fixed 3x3 stencil (0.25 cross) baked into the kernel.
  float* out = (float*)d_out;

  const int nimg = in_sizes[0] / (IMG_H * IMG_W);  // B*C images (C == 1)

  dim3 grid((IMG_W - 2 + TILE_X - 1) / TILE_X,   // 16
            (IMG_H - 2 + TILE_Y - 1) / TILE_Y,   // 32
            nimg);
  jacobi_wmma_kernel<<<grid, dim3(256), 0, stream>>>(u, f, out);

  const int bthreads = nimg * 4 * IMG_W;
  dirichlet_boundary_kernel<<<(bthreads + 255) / 256, 256, 0, stream>>>(out,
                                                                        nimg);
}